// ScatterMax_53429393162897
// MI455X (gfx1250) — compile-verified
//
#include <hip/hip_runtime.h>
#include <stdint.h>

// scatter_max over C channels x N points:
//   out[c,k]    = max(src[c,k], max_{j: idx[j]==k} src[c,j])
//   argmax[c,k] = min j with idx[j]==k and src[c,j]==out[c,k], else N
//
// Memory-bound (no matmul content -> no WMMA). CDNA5-specific paths used:
//   - global_load_async_to_lds_b32 + s_wait_asynccnt (stage index tile, reuse x16)
//   - global_atomic_max_num_f32 (native IEEE f32 max atomic at L2, DEV scope)
//   - b128 non-temporal loads for the streaming src (keep 128MB `out` L2-resident)
//   - ds_load_b128 for staged indices; global_prefetch_b8 for the next channel row

typedef float __attribute__((ext_vector_type(4))) f32x4;
typedef int   __attribute__((ext_vector_type(4))) i32x4;

static constexpr int TILE_J = 1024;  // index entries staged per block (= 256 threads * 4)
static constexpr int CPB    = 16;    // channels processed per block

// -------- pass 1: out = src ; argmax = sentinel(N) --------
__global__ void sm_init_kernel(const float* __restrict__ src,
                               float* __restrict__ out,
                               int* __restrict__ argm,
                               long long total, int sentinel) {
    const long long stride = (long long)gridDim.x * blockDim.x;
    const long long i0     = (long long)blockIdx.x * blockDim.x + threadIdx.x;
    const long long quads  = total >> 2;

    const f32x4* s4 = (const f32x4*)src;
    f32x4*       o4 = (f32x4*)out;
    i32x4*       a4 = (i32x4*)argm;
    i32x4 sv; sv.x = sentinel; sv.y = sentinel; sv.z = sentinel; sv.w = sentinel;

    for (long long q = i0; q < quads; q += stride) {
        f32x4 v = __builtin_nontemporal_load(s4 + q);  // stream src, don't pollute L2
        o4[q] = v;
        a4[q] = sv;
    }
    for (long long i = (quads << 2) + i0; i < total; i += stride) {  // tail (unused for 32M)
        out[i]  = src[i];
        argm[i] = sentinel;
    }
}

// stage low 32 bits of int64 index[j0..j0+TILE_J) into LDS via async DMA
__device__ __forceinline__ void stage_index_tile(const long long* __restrict__ index,
                                                 int j0, int N, int* sidx) {
    for (int t = threadIdx.x; t < TILE_J; t += blockDim.x) {
        int js = j0 + t;
        if (js >= N) js = N - 1;                         // clamp; clamped slots never read
        const void* g = (const char*)index + (size_t)js * 8;  // little-endian low dword
        unsigned l = (unsigned)(uintptr_t)(&sidx[t]);    // low 32 bits = LDS offset
        asm volatile("global_load_async_to_lds_b32 %0, %1, off"
                     :: "v"(l), "v"(g) : "memory");
    }
    asm volatile("s_wait_asynccnt 0" ::: "memory");
    __syncthreads();
}

__device__ __forceinline__ void atomic_max_f32(float* p, float v) {
    asm volatile("global_atomic_max_num_f32 %0, %1, off scope:SCOPE_DEV"
                 :: "v"(p), "v"(v) : "memory");
}

// -------- pass 2: atomic float-max scatter into out --------
__global__ void sm_scatter_kernel(const float* __restrict__ src,
                                  const long long* __restrict__ index,
                                  float* __restrict__ out,
                                  int C, int N) {
    __shared__ int sidx[TILE_J];
    const int j0   = blockIdx.x * TILE_J;
    int tile = N - j0; if (tile > TILE_J) tile = TILE_J;

    stage_index_tile(index, j0, N, sidx);

    const int cbase = blockIdx.y * CPB;
    for (int cc = 0; cc < CPB && (cbase + cc) < C; ++cc) {
        const int c = cbase + cc;
        const float* srcc = src + (size_t)c * N;
        float*       outc = out + (size_t)c * N;
        if (cc + 1 < CPB && c + 1 < C)  // warm next channel row chunk
            __builtin_prefetch(src + (size_t)(c + 1) * N + j0 + (int)threadIdx.x * 4, 0, 0);

        if (tile == TILE_J) {
            // full tile: each thread owns 4 consecutive elements
            const int t = (int)threadIdx.x * 4;
            f32x4 v  = __builtin_nontemporal_load((const f32x4*)(srcc + j0 + t));
            i32x4 kv = *(const i32x4*)(&sidx[t]);        // one ds_load_b128
            atomic_max_f32(outc + kv.x, v.x);
            atomic_max_f32(outc + kv.y, v.y);
            atomic_max_f32(outc + kv.z, v.z);
            atomic_max_f32(outc + kv.w, v.w);
        } else {
            for (int t = threadIdx.x; t < tile; t += blockDim.x) {
                float v = __builtin_nontemporal_load(srcc + j0 + t);
                atomic_max_f32(outc + sidx[t], v);
            }
        }
    }
}

// -------- pass 3: argmax via atomic min where src hits the segment max --------
__global__ void sm_arg_kernel(const float* __restrict__ src,
                              const long long* __restrict__ index,
                              const float* __restrict__ out,
                              int* __restrict__ argm,
                              int C, int N) {
    __shared__ int sidx[TILE_J];
    const int j0   = blockIdx.x * TILE_J;
    int tile = N - j0; if (tile > TILE_J) tile = TILE_J;

    stage_index_tile(index, j0, N, sidx);

    const int cbase = blockIdx.y * CPB;
    for (int cc = 0; cc < CPB && (cbase + cc) < C; ++cc) {
        const int c = cbase + cc;
        const float* srcc = src + (size_t)c * N;
        const float* outc = out + (size_t)c * N;
        int*         argc = argm + (size_t)c * N;
        if (cc + 1 < CPB && c + 1 < C)
            __builtin_prefetch(src + (size_t)(c + 1) * N + j0 + (int)threadIdx.x * 4, 0, 0);

        if (tile == TILE_J) {
            const int t = (int)threadIdx.x * 4;
            f32x4 v  = __builtin_nontemporal_load((const f32x4*)(srcc + j0 + t));
            i32x4 kv = *(const i32x4*)(&sidx[t]);
            // gathers hit L2 (whole `out` fits in 192MB L2)
            if (v.x == outc[kv.x]) atomicMin(argc + kv.x, j0 + t + 0);
            if (v.y == outc[kv.y]) atomicMin(argc + kv.y, j0 + t + 1);
            if (v.z == outc[kv.z]) atomicMin(argc + kv.z, j0 + t + 2);
            if (v.w == outc[kv.w]) atomicMin(argc + kv.w, j0 + t + 3);
        } else {
            for (int t = threadIdx.x; t < tile; t += blockDim.x) {
                float v = __builtin_nontemporal_load(srcc + j0 + t);
                int   k = sidx[t];
                if (v == outc[k])
                    atomicMin(argc + k, j0 + t);
            }
        }
    }
}

extern "C" void kernel_launch(void* const* d_in, const int* in_sizes, int n_in,
                              void* d_out, int out_size, void* d_ws, size_t ws_size,
                              hipStream_t stream) {
    const float*     src   = (const float*)d_in[0];
    const long long* index = (const long long*)d_in[1];

    const int N = in_sizes[1];
    const int C = in_sizes[0] / N;

    float* out  = (float*)d_out;
    int*   argm = (int*)((float*)d_out + (size_t)C * N);

    const long long total = (long long)C * N;
    const int threads = 256;

    // pass 1: seed out with src, argmax with sentinel N
    sm_init_kernel<<<2048, threads, 0, stream>>>(src, out, argm, total, N);

    dim3 grid((N + TILE_J - 1) / TILE_J, (C + CPB - 1) / CPB);

    // pass 2: float-max atomics (gives out = max(src, seg_max))
    sm_scatter_kernel<<<grid, threads, 0, stream>>>(src, index, out, C, N);

    // pass 3: argmin over positions achieving the segment max
    sm_arg_kernel<<<grid, threads, 0, stream>>>(src, index, out, argm, C, N);
}